// HMCLayer_90177133347127
// MI455X (gfx1250) — compile-verified
//
#include <hip/hip_runtime.h>
#include <cstddef>

// ---------------------------------------------------------------------------
// HMC (hierarchical attention message passing), d = 64, f32 end-to-end.
// Dense GEMMs use CDNA5 V_WMMA_F32_16X16X4_F32 (full f32 precision).
// W is staged in LDS in a K-pair-packed layout so every WMMA B fragment is a
// single aligned ds_load_b64 into an even-aligned VGPR pair (no shuffles).
// ---------------------------------------------------------------------------

typedef __attribute__((ext_vector_type(2))) float v2f;
typedef __attribute__((ext_vector_type(8))) float v8f;

#define NEG_SLOPE 0.2f
#define ROW_EPS   1e-9f

// ---------------- utility kernels ----------------

__global__ __launch_bounds__(256) void k_zero(float* __restrict__ p, size_t n) {
  size_t i = (size_t)blockIdx.x * blockDim.x + threadIdx.x;
  if (i < n) p[i] = 0.0f;
}

__global__ __launch_bounds__(256) void k_sigmoid(float* __restrict__ p, size_t n) {
  size_t i = (size_t)blockIdx.x * blockDim.x + threadIdx.x;
  if (i < n) {
    float v = p[i];
    p[i] = 1.0f / (1.0f + __expf(-v));
  }
}

// ---------------- dense GEMM: msg[n,64] = x[n,64] @ W[64,64] ----------------
// 8 wave32 per block, each wave computes a 16-row strip using
// v_wmma_f32_16x16x4_f32 (16 K-steps x 4 column tiles).

__global__ __launch_bounds__(256)
void k_gemm64(const float* __restrict__ x, const float* __restrict__ w,
              float* __restrict__ msg, int n) {
  // K-pair-packed W: wl2[kp*128 + col*2 + j] = W[2*kp + j][col]
  __shared__ __align__(16) float wl2[64 * 64];
  const int tid = threadIdx.x;

  // Stage W (2048 pairs, 8 per thread). Global reads are fully coalesced
  // (consecutive threads -> consecutive columns); each pair stores as one b64.
  {
#pragma unroll
    for (int i = 0; i < 8; ++i) {
      int p   = tid + 256 * i;   // pair index 0..2047
      int kp  = p >> 6;          // 0..31
      int col = p & 63;
      v2f pr = { w[(2 * kp) * 64 + col], w[(2 * kp + 1) * 64 + col] };
      *(v2f*)(wl2 + kp * 128 + col * 2) = pr;
    }
  }
  __syncthreads();

  const int wave = tid >> 5;        // 0..7
  const int lane = tid & 31;
  const int row0 = blockIdx.x * 128 + wave * 16;

  int arow = row0 + (lane & 15);    // A-fragment row this lane supplies
  if (arow >= n) arow = n - 1;      // clamp: loads stay in-bounds, stores guarded
  const int   khalf = (lane >> 4) * 2;     // lanes 16..31 hold K+2/K+3
  const int   col   = lane & 15;
  const float* xrow = x + (size_t)arow * 64;

  v8f c0 = {}, c1 = {}, c2 = {}, c3 = {};

#pragma unroll
  for (int kt = 0; kt < 16; ++kt) {
    const int kb = kt * 4 + khalf;
    const int kp = kb >> 1;
    // A fragment (16x4 f32): lane -> x[row][kb], x[row][kb+1] (one b64 load)
    v2f a = { xrow[kb], xrow[kb + 1] };
    // B fragments (4x16 f32): one aligned b64 LDS load per column tile
    const v2f* wp = (const v2f*)(wl2 + kp * 128);
    v2f b0 = wp[col +  0];
    v2f b1 = wp[col + 16];
    v2f b2 = wp[col + 32];
    v2f b3 = wp[col + 48];
    c0 = __builtin_amdgcn_wmma_f32_16x16x4_f32(false, a, false, b0, (short)0, c0, false, false);
    c1 = __builtin_amdgcn_wmma_f32_16x16x4_f32(false, a, false, b1, (short)0, c1, false, false);
    c2 = __builtin_amdgcn_wmma_f32_16x16x4_f32(false, a, false, b2, (short)0, c2, false, false);
    c3 = __builtin_amdgcn_wmma_f32_16x16x4_f32(false, a, false, b3, (short)0, c3, false, false);
  }

  // C/D layout: VGPR i holds row (lane>>4)*8 + i at column (lane&15).
  const int rbase = row0 + (lane >> 4) * 8;
#pragma unroll
  for (int i = 0; i < 8; ++i) {
    const int r = rbase + i;
    if (r < n) {
      float* mr = msg + (size_t)r * 64 + col;
      mr[0]  = c0[i];
      mr[16] = c1[i];
      mr[32] = c2[i];
      mr[48] = c3[i];
    }
  }
}

// ---------------- per-node dual score dots ----------------
// sc[2*r+0] = msg[r] . a0 ; sc[2*r+1] = msg[r] . a1

__global__ __launch_bounds__(256)
void k_scores(const float* __restrict__ msg, const float* __restrict__ a0,
              const float* __restrict__ a1, float* __restrict__ sc, int n) {
  int r = blockIdx.x * blockDim.x + threadIdx.x;
  if (r >= n) return;
  const float* m = msg + (size_t)r * 64;
  float d0 = 0.0f, d1 = 0.0f;
#pragma unroll
  for (int c = 0; c < 64; ++c) {
    float v = m[c];
    d0 += v * a0[c];
    d1 += v * a1[c];
  }
  sc[2 * r]     = d0;
  sc[2 * r + 1] = d1;
}

// ---------------- edge phase 1 ----------------

__device__ __forceinline__ float lrelu(float v) {
  return v > 0.0f ? v : NEG_SLOPE * v;
}

// within-level: e = lrelu(sc_src[rows] + sc_tgt[cols]); rowsum over rows
__global__ __launch_bounds__(256)
void k_edge_aba(const int* __restrict__ rows, const int* __restrict__ cols,
                const float* __restrict__ sc, float* __restrict__ rsum,
                float* __restrict__ val, int E) {
  int e = blockIdx.x * blockDim.x + threadIdx.x;
  if (e >= E) return;
  int r = rows[e], c = cols[e];
  float v = lrelu(sc[2 * r] + sc[2 * c + 1]);
  val[e] = v;
  atomicAdd(&rsum[r], v);
}

// between-level: both directions in one pass
__global__ __launch_bounds__(256)
void k_edge_abi(const int* __restrict__ rows, const int* __restrict__ cols,
                const float* __restrict__ scS, const float* __restrict__ scT,
                float* __restrict__ rsS, float* __restrict__ rsT,
                float* __restrict__ valA, float* __restrict__ valB, int E) {
  int e = blockIdx.x * blockDim.x + threadIdx.x;
  if (e >= E) return;
  int r = rows[e], c = cols[e];
  float vst = lrelu(scS[2 * r]     + scT[2 * c]);      // a_st_s / a_st_t
  float vts = lrelu(scS[2 * r + 1] + scT[2 * c + 1]);  // a_ts_s / a_ts_t
  valA[e] = vst;
  valB[e] = vts;
  atomicAdd(&rsT[c], vst);   // att_st normalized over target rows
  atomicAdd(&rsS[r], vts);   // att_ts normalized over source rows
}

// ---------------- edge phase 2: weighted scatter (64 threads / edge) ----------------

__global__ __launch_bounds__(256)
void k_agg_aba(const int* __restrict__ rows, const int* __restrict__ cols,
               const float* __restrict__ val, const float* __restrict__ rsum,
               const float* __restrict__ msg, float* __restrict__ agg, int E) {
  size_t t = (size_t)blockIdx.x * blockDim.x + threadIdx.x;
  int e = (int)(t >> 6);
  int j = (int)(t & 63);
  if (e >= E) return;
  int r = rows[e], c = cols[e];
  float att = val[e] / (rsum[r] + ROW_EPS);
  atomicAdd(&agg[(size_t)r * 64 + j], att * msg[(size_t)c * 64 + j]);
}

__global__ __launch_bounds__(256)
void k_agg_abi(const int* __restrict__ rows, const int* __restrict__ cols,
               const float* __restrict__ valA, const float* __restrict__ valB,
               const float* __restrict__ rsS, const float* __restrict__ rsT,
               const float* __restrict__ msgS, const float* __restrict__ msgT,
               float* __restrict__ aggS, float* __restrict__ aggT, int E) {
  size_t t = (size_t)blockIdx.x * blockDim.x + threadIdx.x;
  int e = (int)(t >> 6);
  int j = (int)(t & 63);
  if (e >= E) return;
  int r = rows[e], c = cols[e];
  float ast = valA[e] / (rsT[c] + ROW_EPS);
  float ats = valB[e] / (rsS[r] + ROW_EPS);
  atomicAdd(&aggT[(size_t)c * 64 + j], ast * msgS[(size_t)r * 64 + j]);
  atomicAdd(&aggS[(size_t)r * 64 + j], ats * msgT[(size_t)c * 64 + j]);
}

// ---------------- host-side helpers ----------------

static inline unsigned gb(size_t n, unsigned b) { return (unsigned)((n + b - 1) / b); }

static void run_ccaba(const float* x, const int* rows, const int* cols, int E,
                      const float* a_src, const float* a_tgt, const float* w,
                      int n, float* msg, float* sc, float* val, float* rsum,
                      float* agg, hipStream_t s) {
  k_gemm64 <<<gb((size_t)n, 128), 256, 0, s>>>(x, w, msg, n);
  k_scores <<<gb((size_t)n, 256), 256, 0, s>>>(msg, a_src, a_tgt, sc, n);
  k_zero   <<<gb((size_t)n, 256), 256, 0, s>>>(rsum, (size_t)n);
  k_edge_aba<<<gb((size_t)E, 256), 256, 0, s>>>(rows, cols, sc, rsum, val, E);
  k_agg_aba <<<gb((size_t)E * 64, 256), 256, 0, s>>>(rows, cols, val, rsum, msg, agg, E);
}

static void run_ccabi(const float* xs, const float* xt,
                      const int* rows, const int* cols, int E,
                      const float* a_st_s, const float* a_st_t,
                      const float* a_ts_s, const float* a_ts_t,
                      const float* w_s, const float* w_t,
                      int ns, int nt,
                      float* msgS, float* msgT, float* scS, float* scT,
                      float* valA, float* valB, float* rsS, float* rsT,
                      float* aggS, float* aggT, hipStream_t s) {
  k_gemm64 <<<gb((size_t)ns, 128), 256, 0, s>>>(xs, w_s, msgS, ns);
  k_scores <<<gb((size_t)ns, 256), 256, 0, s>>>(msgS, a_st_s, a_ts_s, scS, ns);
  k_gemm64 <<<gb((size_t)nt, 128), 256, 0, s>>>(xt, w_t, msgT, nt);
  k_scores <<<gb((size_t)nt, 256), 256, 0, s>>>(msgT, a_st_t, a_ts_t, scT, nt);
  k_zero   <<<gb((size_t)ns, 256), 256, 0, s>>>(rsS, (size_t)ns);
  k_zero   <<<gb((size_t)nt, 256), 256, 0, s>>>(rsT, (size_t)nt);
  k_edge_abi<<<gb((size_t)E, 256), 256, 0, s>>>(rows, cols, scS, scT, rsS, rsT, valA, valB, E);
  k_agg_abi <<<gb((size_t)E * 64, 256), 256, 0, s>>>(rows, cols, valA, valB, rsS, rsT,
                                                     msgS, msgT, aggS, aggT, E);
}

// ---------------- entry point ----------------

extern "C" void kernel_launch(void* const* d_in, const int* in_sizes, int n_in,
                              void* d_out, int out_size, void* d_ws, size_t ws_size,
                              hipStream_t stream) {
  (void)n_in; (void)out_size; (void)ws_size;

  const float* x0 = (const float*)d_in[0];
  const float* x1 = (const float*)d_in[1];
  const float* x2 = (const float*)d_in[2];
  const int n0 = in_sizes[0] / 64;
  const int n1 = in_sizes[1] / 64;
  const int n2 = in_sizes[2] / 64;

  const int* adj0 = (const int*)d_in[3]; const int E0  = in_sizes[3] / 2;
  const int* adj1 = (const int*)d_in[4]; const int E1  = in_sizes[4] / 2;
  const int* co2  = (const int*)d_in[5]; const int E2  = in_sizes[5] / 2;
  const int* i1r  = (const int*)d_in[6]; const int Ei1 = in_sizes[6];
  const int* i1c  = (const int*)d_in[7];
  const int* i2r  = (const int*)d_in[8]; const int Ei2 = in_sizes[8];
  const int* i2c  = (const int*)d_in[9];

  // params: pytree (sorted-key) flattening, 36 leaves after the 10 tensors.
  // ccaba -> {a_src, a_tgt, w} ; ccabi -> {a_st_s, a_st_t, a_ts_s, a_ts_t, w_s, w_t}
  // order: c1_00[0..2] c1_01[3..8] c1_12[9..14] c2_00[15..17] c2_01[18..23]
  //        c2_11[24..26] c2_12[27..32] c2_22[33..35]
  const float* const* P = (const float* const*)(d_in + 10);

  // ---- workspace layout ----
  float* ws = (float*)d_ws;
  size_t nmax = (size_t)(n1 > n0 ? (n1 > n2 ? n1 : n2) : (n0 > n2 ? n0 : n2));
  size_t Emax = (size_t)E1;
  if ((size_t)E0 > Emax) Emax = E0;  if ((size_t)E2 > Emax) Emax = E2;
  if ((size_t)Ei1 > Emax) Emax = Ei1; if ((size_t)Ei2 > Emax) Emax = Ei2;

  size_t o = 0;
  float* msgS = ws + o; o += nmax * 64;
  float* msgT = ws + o; o += nmax * 64;
  float* scS  = ws + o; o += nmax * 2;
  float* scT  = ws + o; o += nmax * 2;
  float* valA = ws + o; o += Emax;
  float* valB = ws + o; o += Emax;
  float* rsS  = ws + o; o += nmax;
  float* rsT  = ws + o; o += nmax;
  float* xl0  = ws + o; o += (size_t)n0 * 64;   // level-1 features (contiguous block)
  float* xl1  = ws + o; o += (size_t)n1 * 64;
  float* xl2  = ws + o; o += (size_t)n2 * 64;

  const size_t nfeat = (size_t)(n0 + n1 + n2) * 64;

  // ================= Level 1 =================
  k_zero<<<gb(nfeat, 256), 256, 0, stream>>>(xl0, nfeat);  // xl0/xl1/xl2 contiguous

  // c1_00 : ccaba(x_0, adj0) -> xl0
  run_ccaba(x0, adj0, adj0 + E0, E0, P[0], P[1], P[2],
            n0, msgS, scS, valA, rsS, xl0, stream);
  // c1_01 : ccabi(x_0, x_1) -> (aggT=xl1, aggS=xl0)
  run_ccabi(x0, x1, i1r, i1c, Ei1, P[3], P[4], P[5], P[6], P[7], P[8],
            n0, n1, msgS, msgT, scS, scT, valA, valB, rsS, rsT, xl0, xl1, stream);
  // c1_12 : ccabi(x_1, x_2) -> (aggT=xl2, aggS=xl1)
  run_ccabi(x1, x2, i2r, i2c, Ei2, P[9], P[10], P[11], P[12], P[13], P[14],
            n1, n2, msgS, msgT, scS, scT, valA, valB, rsS, rsT, xl1, xl2, stream);

  k_sigmoid<<<gb(nfeat, 256), 256, 0, stream>>>(xl0, nfeat);  // in place -> level-1 features

  // ================= Level 2 (accumulate straight into d_out) =================
  float* out0 = (float*)d_out;
  float* out1 = out0 + (size_t)n0 * 64;
  float* out2 = out1 + (size_t)n1 * 64;
  k_zero<<<gb(nfeat, 256), 256, 0, stream>>>(out0, nfeat);

  // c2_00 : ccaba(xl0, adj0) -> out0
  run_ccaba(xl0, adj0, adj0 + E0, E0, P[15], P[16], P[17],
            n0, msgS, scS, valA, rsS, out0, stream);
  // c2_01 : ccabi(xl0, xl1) -> (aggT=out1, aggS=out0)
  run_ccabi(xl0, xl1, i1r, i1c, Ei1, P[18], P[19], P[20], P[21], P[22], P[23],
            n0, n1, msgS, msgT, scS, scT, valA, valB, rsS, rsT, out0, out1, stream);
  // c2_11 : ccaba(xl1, adj1) -> out1
  run_ccaba(xl1, adj1, adj1 + E1, E1, P[24], P[25], P[26],
            n1, msgS, scS, valA, rsS, out1, stream);
  // c2_12 : ccabi(xl1, xl2) -> (aggT=out2, aggS=out1)
  run_ccabi(xl1, xl2, i2r, i2c, Ei2, P[27], P[28], P[29], P[30], P[31], P[32],
            n1, n2, msgS, msgT, scS, scT, valA, valB, rsS, rsT, out1, out2, stream);
  // c2_22 : ccaba(xl2, coadj2) -> out2
  run_ccaba(xl2, co2, co2 + E2, E2, P[33], P[34], P[35],
            n2, msgS, scS, valA, rsS, out2, stream);

  k_sigmoid<<<gb(nfeat, 256), 256, 0, stream>>>(out0, nfeat);
}